// NeuralTuringMachine_87153476370956
// MI455X (gfx1250) — compile-verified
//
#include <hip/hip_runtime.h>
#include <hip/hip_bf16.h>
#include <math.h>

// ---------------- problem constants ----------------
#define ROWS 32768
#define COLS 1024
#define D_IN 256
#define D_OUT 256
#define D_CTRL 1024
#define D_CTRL_IN (D_IN + COLS)            // 1280
#define KSHIFT 3
#define HEADSZ (COLS + 1 + 1 + KSHIFT + 1) // 1030
#define N_PARAMS (2 * HEADSZ + 2 * COLS)   // 4108

// ---------------- output layout (flat, return order) ----------------
static constexpr size_t OUT_SAIDA  = 0;
static constexpr size_t OUT_MEM    = D_OUT;                       // 256
static constexpr size_t OUT_RVEC   = OUT_MEM + (size_t)ROWS*COLS; // 33554688
static constexpr size_t OUT_WREAD  = OUT_RVEC + COLS;             // 33555712
static constexpr size_t OUT_WWRITE = OUT_WREAD + ROWS;            // 33588480

// ---------------- workspace layout (floats) ----------------
#define SIM_RB 128   // rows per block in k_sim
#define UPD_RPB 128  // rows per block in k_update
static constexpr size_t WS_H    = 0;
static constexpr size_t WS_P    = WS_H + D_CTRL;        // 1024
static constexpr size_t WS_SCAL = WS_P + N_PARAMS;      // 5132 (16 slots, stride 8/head)
static constexpr size_t WS_KR   = WS_SCAL + 16;
static constexpr size_t WS_KW   = WS_KR + COLS;
static constexpr size_t WS_E    = WS_KW + COLS;
static constexpr size_t WS_A    = WS_E + COLS;
static constexpr size_t WS_SIMR = WS_A + COLS;
static constexpr size_t WS_SIMW = WS_SIMR + ROWS;
static constexpr size_t WS_WG   = WS_SIMW + ROWS;        // 2*ROWS
static constexpr size_t WS_PART = WS_WG + 2*(size_t)ROWS; // (ROWS/UPD_RPB)*COLS

typedef __attribute__((ext_vector_type(2))) float v2f;
typedef __attribute__((ext_vector_type(8))) float v8f;
typedef __attribute__((ext_vector_type(4))) unsigned int u32x4;
typedef __attribute__((ext_vector_type(8))) int i32x8;
typedef __attribute__((ext_vector_type(4))) int i32x4;

#if defined(__has_builtin)
#if __has_builtin(__builtin_amdgcn_tensor_load_to_lds) && \
    __has_builtin(__builtin_amdgcn_s_wait_tensorcnt)
#define USE_TDM 1
#endif
#endif
#ifndef USE_TDM
#define USE_TDM 0
#endif

__device__ __forceinline__ float sigmoidf(float x) { return 1.f / (1.f + __expf(-x)); }
__device__ __forceinline__ float softplusf(float x) {
  return (x > 20.f) ? x : log1pf(__expf(x));
}

// ---------------- kernel 1: h = Wc @ [x; read_prev] + bc (wave per row) ----------------
__global__ void __launch_bounds__(256) k_controller_h(
    const float* __restrict__ x, const float* __restrict__ read_prev,
    const float* __restrict__ Wc, const float* __restrict__ bc,
    float* __restrict__ h) {
  int tid = blockIdx.x * blockDim.x + threadIdx.x;
  int row = tid >> 5, lane = tid & 31;
  if (row >= D_CTRL) return;
  const float* wrow = Wc + (size_t)row * D_CTRL_IN;
  float acc = 0.f;
  for (int j = lane; j < D_CTRL_IN; j += 32) {
    float xv = (j < D_IN) ? x[j] : read_prev[j - D_IN];
    acc = fmaf(wrow[j], xv, acc);
  }
  for (int o = 16; o; o >>= 1) acc += __shfl_down(acc, o, 32);
  if (lane == 0) h[row] = acc + bc[row];
}

// ---------------- kernel 2: p = Wp@h + bp ; saida = sigmoid(Wo@h + bo) ----------------
__global__ void __launch_bounds__(256) k_controller_p(
    const float* __restrict__ h,
    const float* __restrict__ Wp, const float* __restrict__ bp,
    const float* __restrict__ Wo, const float* __restrict__ bo,
    float* __restrict__ p, float* __restrict__ saida) {
  int tid = blockIdx.x * blockDim.x + threadIdx.x;
  int row = tid >> 5, lane = tid & 31;
  if (row >= N_PARAMS + D_OUT) return;
  const float* wrow = (row < N_PARAMS) ? (Wp + (size_t)row * D_CTRL)
                                       : (Wo + (size_t)(row - N_PARAMS) * D_CTRL);
  float acc = 0.f;
  for (int j = lane; j < D_CTRL; j += 32) acc = fmaf(wrow[j], h[j], acc);
  for (int o = 16; o; o >>= 1) acc += __shfl_down(acc, o, 32);
  if (lane == 0) {
    if (row < N_PARAMS) p[row] = acc + bp[row];
    else saida[row - N_PARAMS] = sigmoidf(acc + bo[row - N_PARAMS]);
  }
}

// ---------------- kernel 3: head params + norms ----------------
// scal layout (stride 8 per head): [beta, g, s0, s1, s2, gamma, inv_knorm, -]
__global__ void __launch_bounds__(1024) k_params(
    const float* __restrict__ p,
    float* __restrict__ kr, float* __restrict__ kw,
    float* __restrict__ ev, float* __restrict__ av,
    float* __restrict__ scal) {
  __shared__ float red[1024];
  int t = threadIdx.x;
  float krv = tanhf(p[t]);
  float kwv = tanhf(p[HEADSZ + t]);
  kr[t] = krv;
  kw[t] = kwv;
  ev[t] = sigmoidf(p[2 * HEADSZ + t]);
  av[t] = tanhf(p[2 * HEADSZ + COLS + t]);

  // ||kr||
  red[t] = krv * krv; __syncthreads();
  for (int s = 512; s; s >>= 1) { if (t < s) red[t] += red[t + s]; __syncthreads(); }
  if (t == 0) scal[6] = 1.f / fmaxf(sqrtf(red[0]), 1e-12f);
  __syncthreads();
  // ||kw||
  red[t] = kwv * kwv; __syncthreads();
  for (int s = 512; s; s >>= 1) { if (t < s) red[t] += red[t + s]; __syncthreads(); }
  if (t == 0) scal[8 + 6] = 1.f / fmaxf(sqrtf(red[0]), 1e-12f);

  if (t < 2) {  // t=0 read head, t=1 write head
    int base = t * HEADSZ;
    float* sc = scal + t * 8;
    sc[0] = softplusf(p[base + COLS]);
    sc[1] = sigmoidf(p[base + COLS + 1]);
    float l0 = p[base + COLS + 2], l1 = p[base + COLS + 3], l2 = p[base + COLS + 4];
    float mx = fmaxf(l0, fmaxf(l1, l2));
    float e0 = __expf(l0 - mx), e1 = __expf(l1 - mx), e2 = __expf(l2 - mx);
    float zi = 1.f / (e0 + e1 + e2);
    sc[2] = e0 * zi; sc[3] = e1 * zi; sc[4] = e2 * zi;
    sc[5] = softplusf(p[base + COLS + 2 + KSHIFT]) + 1.f;
  }
}

// ---------------- kernel 4: cosine similarities via V_WMMA_F32_16X16X4_F32 ----------------
// Each 256-thread block handles 128 memory rows; wave w owns rows [w*16, w*16+16).
// Tile feed: TDM (tensor_load_to_lds) double-buffered when available, else
// cooperative b128 loads. 128x32 fp32 tiles, LDS row stride 36 floats
// (= 32 data DWORDs + 4 pad DWORDs; matches TDM pad_interval=32dw, pad_amount=4dw).
#define LDS_STRIDE 36
#define SIM_TILE_F (SIM_RB * LDS_STRIDE)

#if USE_TDM
__device__ __forceinline__ void tdm_load_tile(const float* gsrc, unsigned lds_off) {
  unsigned long long ga = (unsigned long long)(size_t)gsrc;
  // D# group 0: count=1 | lds_addr | global_addr[56:0] | type=2
  u32x4 g0 = { 1u,
               lds_off,
               (unsigned)ga,
               (unsigned)(ga >> 32) | (2u << 30) };
  // D# group 1:
  // w0: wg_mask=0, data_size=2(4B), pad_enable=1, pad_interval=4(32dw), pad_amount=3(4dw)
  // tensor_dim0=1024, tensor_dim1=32768, tile_dim0=32, tile_dim1=128,
  // tensor_dim0_stride=1024 elements
  i32x8 g1 = { (int)((2u << 16) | (1u << 20) | (4u << 22) | (3u << 25)),
               (int)(((unsigned)COLS & 0xFFFFu) << 16), // tensor_dim0 lo16 -> bits 63:48
               (int)0x80000000u,                        // dim0 hi=0 | dim1 lo16 (32768)
               (int)(32u << 16),                        // dim1 hi=0 | tile_dim0=32
               (int)128,                                // tile_dim1=128 | tile_dim2=0
               (int)COLS,                               // tensor_dim0_stride lo32
               0, 0 };
  i32x4 gz4 = { 0, 0, 0, 0 };
  i32x8 gz8 = { 0, 0, 0, 0, 0, 0, 0, 0 };
  // 6-arg variant (clang-23 / therock-10.0): groups 0..3 + extra group + cpol
  __builtin_amdgcn_tensor_load_to_lds(g0, g1, gz4, gz4, gz8, 0);
}
#endif

__global__ void __launch_bounds__(256) k_sim(
    const float* __restrict__ mem,
    const float* __restrict__ kr, const float* __restrict__ kw,
    const float* __restrict__ scal,
    float* __restrict__ simr, float* __restrict__ simw) {
  __shared__ float tileA[SIM_TILE_F];          // 18.4 KB
  __shared__ float tileB[SIM_TILE_F];          // 18.4 KB (TDM ping-pong; fallback uses A only)
  __shared__ float krs[COLS];                  // 4 KB
  __shared__ float kws[COLS];                  // 4 KB
  __shared__ float simtile[8][16][2];          // 1 KB

  int tid = threadIdx.x;
  int w = tid >> 5, L = tid & 31;
  int n = L & 15, hb = L >> 4;  // lane's N-column / K-half
  int row0 = blockIdx.x * SIM_RB;

  // stage key vectors once
  for (int it = 0; it < 4; ++it) {
    int idx = tid + it * 256;
    krs[idx] = kr[idx];
    kws[idx] = kw[idx];
  }
  __syncthreads();

  // branch-free B operand source: pointer select + 0/1 mask (keeps EXEC all-ones)
  const float* bptr = (n == 1) ? kws : krs;
  float bmask = (n < 2) ? 1.f : 0.f;

  v8f c = {};
  float nrm = 0.f;

#if USE_TDM
  if (w == 0) {
    tdm_load_tile(mem + (size_t)row0 * COLS, (unsigned)(size_t)(void*)tileA);
    __builtin_amdgcn_s_wait_tensorcnt(0);
  }
  __syncthreads();
#endif

  int ib = 0;
  for (int kc = 0; kc < COLS; kc += 32, ib ^= 1) {
#if USE_TDM
    const float* tb = ib ? tileB : tileA;
    float* tnext = ib ? tileA : tileB;
    bool more = (kc + 32) < COLS;
    if (w == 0 && more)
      tdm_load_tile(mem + (size_t)row0 * COLS + kc + 32,
                    (unsigned)(size_t)(void*)tnext);
#else
    const float* tb = tileA;
    // coalesced global -> LDS: 128 rows x 32 cols, float4 per thread x4
#pragma unroll
    for (int it = 0; it < 4; ++it) {
      int l = tid + it * 256;
      int r = l >> 3, c4 = (l & 7) << 2;
      float4 v4 = *(const float4*)(mem + (size_t)(row0 + r) * COLS + kc + c4);
      *(float4*)&tileA[r * LDS_STRIDE + c4] = v4;
    }
    __syncthreads();
#endif

    // row-norm partials: lane handles its row's K-half (16 values)
    {
      const float* rp = &tb[(w * 16 + n) * LDS_STRIDE + hb * 16];
#pragma unroll
      for (int j = 0; j < 16; ++j) { float q = rp[j]; nrm = fmaf(q, q, nrm); }
    }

    // 8x WMMA f32 16x16x4 over this 32-wide K chunk
#pragma unroll
    for (int kk = 0; kk < 8; ++kk) {
      // A 16x4 layout: lanes 0-15 hold K=0,1 (VGPR0,1); lanes 16-31 hold K=2,3
      const float* ap = &tb[(w * 16 + n) * LDS_STRIDE + kk * 4 + 2 * hb];
      v2f a; a.x = ap[0]; a.y = ap[1];
      // B 4x16 mirrored layout: this lane supplies B[2*hb][n], B[2*hb+1][n]
      int col = kc + kk * 4 + 2 * hb;
      v2f b;
      b.x = bptr[col] * bmask;
      b.y = bptr[col + 1] * bmask;
      c = __builtin_amdgcn_wmma_f32_16x16x4_f32(false, a, false, b, (short)0, c,
                                                false, false);
    }

#if USE_TDM
    if (w == 0 && more) __builtin_amdgcn_s_wait_tensorcnt(0);
#endif
    __syncthreads();
  }

  // combine the two K-half norm partials per row
  nrm += __shfl_xor(nrm, 16, 32);

  // extract D columns 0 (kr dot) and 1 (kw dot): lane L vgpr vi -> M = vi + 8*hb, N = n
  if (n < 2) {
#pragma unroll
    for (int vi = 0; vi < 8; ++vi) simtile[w][vi + 8 * hb][n] = c[vi];
  }
  __syncthreads();

  if (L < 16) {
    float inv_row = 1.f / fmaxf(sqrtf(nrm), 1e-12f);
    int row = row0 + w * 16 + L;
    simr[row] = simtile[w][L][0] * inv_row * scal[6];
    simw[row] = simtile[w][L][1] * inv_row * scal[8 + 6];
  }
}

// ---------------- kernel 5: softmax + gate + circular shift + sharpen ----------------
__global__ void __launch_bounds__(1024) k_addr(
    const float* __restrict__ simr, const float* __restrict__ simw,
    const float* __restrict__ wrp, const float* __restrict__ wwp,
    const float* __restrict__ scal, float* __restrict__ wg_ws,
    float* __restrict__ wread_out, float* __restrict__ wwrite_out) {
  int head = blockIdx.x;
  const float* sim   = head ? simw : simr;
  const float* wprev = head ? wwp : wrp;
  float* wout        = head ? wwrite_out : wread_out;
  const float* sc    = scal + head * 8;
  float* wg          = wg_ws + (size_t)head * ROWS;

  int t = threadIdx.x;
  __shared__ float red[1024];
  float beta = sc[0], g = sc[1], s0 = sc[2], s1 = sc[3], s2 = sc[4], gam = sc[5];

  // max(beta*sim)
  float m = -1e30f;
  for (int i = t; i < ROWS; i += 1024) m = fmaxf(m, beta * sim[i]);
  red[t] = m; __syncthreads();
  for (int s = 512; s; s >>= 1) { if (t < s) red[t] = fmaxf(red[t], red[t + s]); __syncthreads(); }
  m = red[0]; __syncthreads();

  // sum exp
  float z = 0.f;
  for (int i = t; i < ROWS; i += 1024) z += __expf(beta * sim[i] - m);
  red[t] = z; __syncthreads();
  for (int s = 512; s; s >>= 1) { if (t < s) red[t] += red[t + s]; __syncthreads(); }
  float invz = 1.f / red[0]; __syncthreads();

  // w_g = g*softmax + (1-g)*w_prev  -> ws (needed for neighbor taps)
  for (int i = t; i < ROWS; i += 1024) {
    float wc = __expf(beta * sim[i] - m) * invz;
    wg[i] = g * wc + (1.f - g) * wprev[i];
  }
  __threadfence();
  __syncthreads();

  // circular 3-tap shift + sharpening
  float wp[ROWS / 1024];
  float ssum = 0.f;
#pragma unroll 4
  for (int j = 0; j < ROWS / 1024; ++j) {
    int i = t + j * 1024;
    int im = (i == 0) ? ROWS - 1 : i - 1;
    int ip = (i == ROWS - 1) ? 0 : i + 1;
    float wt = s0 * wg[im] + s1 * wg[i] + s2 * wg[ip];
    float v = powf(fmaxf(wt, 0.f), gam);
    wp[j] = v;
    ssum += v;
  }
  red[t] = ssum; __syncthreads();
  for (int s = 512; s; s >>= 1) { if (t < s) red[t] += red[t + s]; __syncthreads(); }
  float inv = 1.f / (red[0] + 1e-8f);
#pragma unroll 4
  for (int j = 0; j < ROWS / 1024; ++j) wout[t + j * 1024] = wp[j] * inv;
}

// ---------------- kernel 6: fused read_vec partials + erase/add write ----------------
// 256 threads x 4 cols (float4) per block; 128 rows per block.
__global__ void __launch_bounds__(256) k_update(
    const float* __restrict__ mem, const float* __restrict__ wread,
    const float* __restrict__ wwrite, const float* __restrict__ ev,
    const float* __restrict__ av, float* __restrict__ mem_new,
    float* __restrict__ part) {
  int c0 = threadIdx.x * 4;
  int i0 = blockIdx.x * UPD_RPB;
  float4 e4 = *(const float4*)(ev + c0);
  float4 a4 = *(const float4*)(av + c0);
  float4 acc = make_float4(0.f, 0.f, 0.f, 0.f);
  for (int i = i0; i < i0 + UPD_RPB; ++i) {
    size_t off = (size_t)i * COLS + c0;
    __builtin_prefetch(mem + off + (size_t)8 * COLS, 0, 0);
    float4 m4 = *(const float4*)(mem + off);
    float wri = wread[i], wwi = wwrite[i];
    acc.x = fmaf(wri, m4.x, acc.x);
    acc.y = fmaf(wri, m4.y, acc.y);
    acc.z = fmaf(wri, m4.z, acc.z);
    acc.w = fmaf(wri, m4.w, acc.w);
    float4 o4;
    o4.x = m4.x * (1.f - wwi * e4.x) + wwi * a4.x;
    o4.y = m4.y * (1.f - wwi * e4.y) + wwi * a4.y;
    o4.z = m4.z * (1.f - wwi * e4.z) + wwi * a4.z;
    o4.w = m4.w * (1.f - wwi * e4.w) + wwi * a4.w;
    *(float4*)(mem_new + off) = o4;
  }
  *(float4*)(part + (size_t)blockIdx.x * COLS + c0) = acc;
}

// ---------------- kernel 7: deterministic read_vec reduction ----------------
__global__ void __launch_bounds__(1024) k_rvec(const float* __restrict__ part,
                                               float* __restrict__ rvec) {
  int c = threadIdx.x;
  float s = 0.f;
  for (int b = 0; b < ROWS / UPD_RPB; ++b) s += part[(size_t)b * COLS + c];
  rvec[c] = s;
}

// ---------------- host launch ----------------
extern "C" void kernel_launch(void* const* d_in, const int* in_sizes, int n_in,
                              void* d_out, int out_size, void* d_ws, size_t ws_size,
                              hipStream_t stream) {
  const float* x         = (const float*)d_in[0];
  const float* memoria   = (const float*)d_in[1];
  const float* Wc        = (const float*)d_in[2];
  const float* bc        = (const float*)d_in[3];
  const float* Wp        = (const float*)d_in[4];
  const float* bp        = (const float*)d_in[5];
  const float* Wo        = (const float*)d_in[6];
  const float* bo        = (const float*)d_in[7];
  const float* read_prev = (const float*)d_in[8];
  const float* wrp       = (const float*)d_in[9];
  const float* wwp       = (const float*)d_in[10];

  float* out     = (float*)d_out;
  float* saida   = out + OUT_SAIDA;
  float* mem_new = out + OUT_MEM;
  float* rvec    = out + OUT_RVEC;
  float* wread   = out + OUT_WREAD;
  float* wwrite  = out + OUT_WWRITE;

  float* ws   = (float*)d_ws;
  float* h    = ws + WS_H;
  float* p    = ws + WS_P;
  float* scal = ws + WS_SCAL;
  float* kr   = ws + WS_KR;
  float* kw   = ws + WS_KW;
  float* ev   = ws + WS_E;
  float* av   = ws + WS_A;
  float* simr = ws + WS_SIMR;
  float* simw = ws + WS_SIMW;
  float* wg   = ws + WS_WG;
  float* part = ws + WS_PART;

  k_controller_h<<<(D_CTRL * 32 + 255) / 256, 256, 0, stream>>>(x, read_prev, Wc, bc, h);
  k_controller_p<<<((N_PARAMS + D_OUT) * 32 + 255) / 256, 256, 0, stream>>>(
      h, Wp, bp, Wo, bo, p, saida);
  k_params<<<1, 1024, 0, stream>>>(p, kr, kw, ev, av, scal);
  k_sim<<<ROWS / SIM_RB, 256, 0, stream>>>(memoria, kr, kw, scal, simr, simw);
  k_addr<<<2, 1024, 0, stream>>>(simr, simw, wrp, wwp, scal, wg, wread, wwrite);
  k_update<<<ROWS / UPD_RPB, 256, 0, stream>>>(memoria, wread, wwrite, ev, av,
                                               mem_new, part);
  k_rvec<<<1, 1024, 0, stream>>>(part, rvec);
}